// VIGNetFP_9749575762887
// MI455X (gfx1250) — compile-verified
//
#include <hip/hip_runtime.h>
#include <math.h>

// ---------------------------------------------------------------------------
// VIGNet fused inference for MI455X (gfx1250, wave32).
// One workgroup per batch element; all intermediates LDS-resident (~245 KB of
// the 320 KB/WGP). Heavy 10->10ch 5-tap convs and the (2,1)x20 conv run as
// V_WMMA_F32_16X16X4_F32 GEMM tiles (M=16 timesteps, N=16 chans, K padded).
// Attention (MHRSSA) / conv1 / softmax run on VALU. Single HBM pass: read x
// once (8 MB), write 4 KB -> far under the 23.3 TB/s roofline.
//
// v2: A-fragment LDS reads are unconditional aligned v2f (ds_load_b64);
// K-padding correctness comes from zeroed B fragments + zeroed LDS pads,
// so no EXEC-mask branches remain inside the WMMA loops.
// ---------------------------------------------------------------------------

#define TMAX 1008      // padded time extent (real T = 1000)
#define NT   256       // 8 wave32 per workgroup
#define EPSC 1e-3f
#define ALPHA 0.2f

typedef __attribute__((ext_vector_type(2))) float v2f;
typedef __attribute__((ext_vector_type(8))) float v8f;

struct Params { const float* p[35]; };

extern __shared__ float smem[];

__device__ __forceinline__ float lrelu(float x) { return x >= 0.f ? x : ALPHA * x; }

__device__ __forceinline__ v8f wmma_f32(v2f a, v2f b, v8f c) {
  // D = A(16x4) * B(4x16) + C(16x16), full fp32
  return __builtin_amdgcn_wmma_f32_16x16x4_f32(false, a, false, b, (short)0, c,
                                               false, false);
}

// ---------------------------------------------------------------------------
// MHRSSA: m[i][t][o] = sum_{h,c} in[h][t][c]*K[i,h,c,o] + b[i,o]
//         y[o][t][i] = sum_w m[i][t+w][o]*dw[w,i] + db[i]; softmax over i
// mscr reuses the next conv's destination buffer (overwritten afterwards).
// ---------------------------------------------------------------------------
__device__ void att_block(const float* in, int Cin, int Tt,
                          const float* K, const float* bvec,
                          const float* dw, const float* db,
                          float* mscr, float* attb, int tid)
{
  int total = 10 * Tt * 2;
  for (int e = tid; e < total; e += NT) {
    int i = e / (Tt * 2);
    int r = e - i * Tt * 2;
    int t = r >> 1;
    int o = r & 1;
    float acc = bvec[i * 2 + o];
    for (int h2 = 0; h2 < 2; ++h2)
      for (int c = 0; c < Cin; ++c)
        acc += in[(h2 * TMAX + t) * Cin + c] * K[((i * 2 + h2) * Cin + c) * 2 + o];
    mscr[(i * TMAX + t) * 2 + o] = acc;
  }
  __syncthreads();

  int Tt4 = Tt - 4;
  for (int e = tid; e < 2 * Tt4; e += NT) {
    int t = e % Tt4;
    int o = e / Tt4;
    float y[10];
    float mx = -1e30f;
    for (int i = 0; i < 10; ++i) {
      float acc = db[i];
      for (int w = 0; w < 5; ++w)
        acc += mscr[(i * TMAX + t + w) * 2 + o] * dw[w * 10 + i];
      y[i] = acc;
      mx = fmaxf(mx, acc);
    }
    float s = 0.f;
    for (int i = 0; i < 10; ++i) { y[i] = __expf(y[i] - mx); s += y[i]; }
    float inv = 1.f / s;
    for (int i = 0; i < 10; ++i)
      attb[(o * TMAX + t) * 10 + i] = y[i] * inv;
  }
  __syncthreads();
}

// ---------------------------------------------------------------------------
// conv1: 1ch -> 10ch, 5 taps (K=5: too small for WMMA, scalar VALU).
// Fused BN(inference) + leaky-relu + attention mask. Zeroes pad region.
// ---------------------------------------------------------------------------
__device__ void conv1_block(const float* xb, float* out, const float* att,
                            const float* kk, const float* bias,
                            const float* g, const float* be,
                            const float* mu, const float* vv, int tid)
{
  const int Tout = 996;
  for (int e = tid; e < 2 * TMAX * 10; e += NT) {
    int co = e % 10;
    int r = e / 10;
    int t = r % TMAX;
    int h = r / TMAX;
    float val = 0.f;
    if (t < Tout) {
      float acc = bias[co];
      for (int w = 0; w < 5; ++w)
        acc += xb[h * TMAX + t + w] * kk[w * 10 + co];
      float sc = g[co] * rsqrtf(vv[co] + EPSC);
      val = lrelu((acc - mu[co]) * sc + be[co]) * att[(h * TMAX + t) * 10 + co];
    }
    out[(h * TMAX + t) * 10 + co] = val;
  }
  __syncthreads();
}

// ---------------------------------------------------------------------------
// conv2/conv3: 10ch -> 10ch, 5 taps == GEMM, K = 5*10 = 50 (13 K4-blocks).
// LDS layout [h][t][10] makes A[m][k] a contiguous read in[(t0+m)*10 + k];
// each lane's (k0,k0+1) pair is one aligned ds_load_b64. k=50/51 reads spill
// into the (zeroed) t+5 row and are annihilated by zero B fragments — no
// EXEC-divergent guards in the WMMA loop.
// ---------------------------------------------------------------------------
__device__ void conv_wmma10(const float* in, float* out, const float* att,
                            const float* kk, const float* bias,
                            const float* g, const float* be,
                            const float* mu, const float* vv,
                            int Tout, int tid)
{
  int lane = tid & 31;
  int wave = tid >> 5;
  int n = lane & 15;
  int koff = (lane >> 4) * 2;   // A/B frag K-offset within a K4 block

  float bias_n = 0.f, m_n = 0.f, sc_n = 0.f, be_n = 0.f;
  if (n < 10) {
    bias_n = bias[n];
    m_n = mu[n];
    sc_n = g[n] * rsqrtf(vv[n] + EPSC);
    be_n = be[n];
  }

  // B fragments: B[k][n] = kk[w*100 + ci*10 + n], k = w*10+ci, pad k>=50 / n>=10
  v2f bf[13];
  for (int kb = 0; kb < 13; ++kb) {
    int k0 = kb * 4 + koff;
    float b0 = 0.f, b1 = 0.f;
    if (n < 10) {
      if (k0 < 50)     b0 = kk[(k0 / 10) * 100 + (k0 % 10) * 10 + n];
      if (k0 + 1 < 50) b1 = kk[((k0 + 1) / 10) * 100 + ((k0 + 1) % 10) * 10 + n];
    }
    bf[kb].x = b0; bf[kb].y = b1;
  }

  int tilesT = (Tout + 15) >> 4;
  int mrow = lane & 15;
  for (int tile = wave; tile < 2 * tilesT; tile += NT / 32) {
    int hh = tile / tilesT;
    int t0 = (tile - hh * tilesT) << 4;
    const float* inb = in + (hh * TMAX + t0 + mrow) * 10;   // A row base
    v8f acc = {};
#pragma unroll
    for (int kb = 0; kb < 13; ++kb) {
      int k0 = kb * 4 + koff;
      v2f a = *(const v2f*)(inb + k0);   // aligned: base = t*40B, k0 even
      acc = wmma_f32(a, bf[kb], acc);
    }
    int mbase = (lane < 16) ? 0 : 8;
    for (int v = 0; v < 8; ++v) {
      int t = t0 + mbase + v;
      if (t < Tout && n < 10) {
        float val = lrelu((acc[v] + bias_n - m_n) * sc_n + be_n);
        val *= att[(hh * TMAX + t) * 10 + n];
        out[(hh * TMAX + t) * 10 + n] = val;
      }
    }
  }

  // zero the pad region [Tout, TMAX) so tail-tile A reads stay clean
  int padT = TMAX - Tout;
  for (int e = tid; e < 2 * padT * 10; e += NT) {
    int c = e % 10;
    int r = e / 10;
    int t = Tout + (r % padT);
    int h = r / padT;
    out[(h * TMAX + t) * 10 + c] = 0.f;
  }
  __syncthreads();
}

// ---------------------------------------------------------------------------
// conv4: (2,1) conv, 10ch -> 20ch == GEMM over K = 2*10 = 20 (5 K4-blocks),
// two N-tiles (co 0..15 / 16..19). Output stored dense as [t][20] for the head.
// K-pairs start at even k, so each (k0,k0+1) pair never straddles the
// hh boundary (k=9->10) and is a contiguous aligned v2f LDS read.
// ---------------------------------------------------------------------------
__device__ void conv4_block(const float* in, float* out,
                            const float* kk, const float* bias, int tid)
{
  const int Tout = 988;
  int lane = tid & 31;
  int wave = tid >> 5;
  int n = lane & 15;
  int koff = (lane >> 4) * 2;
  int tilesT = (Tout + 15) >> 4;
  int mrow = lane & 15;

  for (int tile = wave; tile < 2 * tilesT; tile += NT / 32) {
    int nt = tile / tilesT;
    int t0 = (tile - nt * tilesT) << 4;
    int co = nt * 16 + n;

    v2f bf[5];
    for (int kb = 0; kb < 5; ++kb) {
      int k0 = kb * 4 + koff;
      float b0 = 0.f, b1 = 0.f;
      if (co < 20) {
        b0 = kk[(k0 / 10) * 200 + (k0 % 10) * 20 + co];
        b1 = kk[((k0 + 1) / 10) * 200 + ((k0 + 1) % 10) * 20 + co];
      }
      bf[kb].x = b0; bf[kb].y = b1;
    }
    float bias_co = (co < 20) ? bias[co] : 0.f;

    v8f acc = {};
#pragma unroll
    for (int kb = 0; kb < 5; ++kb) {
      int k0 = kb * 4 + koff;
      const float* ap = in + ((k0 / 10) * TMAX + t0 + mrow) * 10 + (k0 % 10);
      v2f a = *(const v2f*)ap;           // aligned, contiguous within pair
      acc = wmma_f32(a, bf[kb], acc);
    }
    int mbase = (lane < 16) ? 0 : 8;
    for (int v = 0; v < 8; ++v) {
      int t = t0 + mbase + v;
      if (t < Tout && co < 20)
        out[t * 20 + co] = lrelu(acc[v] + bias_co);
    }
  }
  __syncthreads();
}

// ---------------------------------------------------------------------------
// Dense head: dot of 19760 elements + bias -> out[b]
// ---------------------------------------------------------------------------
__device__ void dense_block(const float* dbuf, const float* w, const float* bsc,
                            float* red, float* out, int b, int tid)
{
  float p = 0.f;
  for (int e = tid; e < 988 * 20; e += NT)
    p += dbuf[e] * w[e];
  red[tid] = p;
  __syncthreads();
  for (int s = NT / 2; s > 0; s >>= 1) {
    if (tid < s) red[tid] += red[tid + s];
    __syncthreads();
  }
  if (tid == 0) out[b] = red[0] + bsc[0];
}

// ---------------------------------------------------------------------------
__global__ __launch_bounds__(NT) void vignet_kernel(Params P, float* __restrict__ out)
{
  const int b = blockIdx.x;
  const int tid = threadIdx.x;

  float* xbuf = smem;                     // [2][TMAX]
  float* bufA = xbuf + 2 * TMAX;          // [2][TMAX][10]
  float* bufB = bufA + 2 * TMAX * 10;     // [2][TMAX][10]
  float* attb = bufB + 2 * TMAX * 10;     // [2][TMAX][10]
  float* red  = attb + 2 * TMAX * 10;     // [NT]

  const float* x = P.p[0];

  // warm the L2 chain for this block's input rows (global_prefetch_b8)
  __builtin_prefetch(x + (size_t)(b * 2) * 1000 + tid * 16, 0, 1);

  for (int e = tid; e < 2 * TMAX; e += NT) {
    int h = e / TMAX, t = e % TMAX;
    xbuf[e] = (t < 1000) ? x[(b * 2 + h) * 1000 + t] : 0.f;
  }
  __syncthreads();

  // block 1 (att1 m-scratch = bufA, conv1 -> bufA)
  att_block(xbuf, 1, 1000, P.p[23], P.p[24], P.p[25], P.p[26], bufA, attb, tid);
  conv1_block(xbuf, bufA, attb, P.p[1], P.p[2],
              P.p[9], P.p[10], P.p[11], P.p[12], tid);

  // block 2 (att2 m-scratch = bufB, conv2: bufA -> bufB, Tout = 992)
  att_block(bufA, 10, 996, P.p[27], P.p[28], P.p[29], P.p[30], bufB, attb, tid);
  conv_wmma10(bufA, bufB, attb, P.p[3], P.p[4],
              P.p[13], P.p[14], P.p[15], P.p[16], 992, tid);

  // block 3 (att3 m-scratch = bufA, conv3: bufB -> bufA, Tout = 988)
  att_block(bufB, 10, 992, P.p[31], P.p[32], P.p[33], P.p[34], bufA, attb, tid);
  conv_wmma10(bufB, bufA, attb, P.p[5], P.p[6],
              P.p[17], P.p[18], P.p[19], P.p[20], 988, tid);

  // conv4: bufA -> bufB (as [988][20]), then dense head
  conv4_block(bufA, bufB, P.p[7], P.p[8], tid);
  dense_block(bufB, P.p[21], P.p[22], red, out, b, tid);
}

extern "C" void kernel_launch(void* const* d_in, const int* in_sizes, int n_in,
                              void* d_out, int out_size, void* d_ws, size_t ws_size,
                              hipStream_t stream)
{
  (void)in_sizes; (void)n_in; (void)d_ws; (void)ws_size; (void)out_size;
  Params P;
  for (int i = 0; i < 35; ++i) P.p[i] = (const float*)d_in[i];
  size_t shmem = (size_t)(2 * TMAX + 3 * (2 * TMAX * 10) + NT) * sizeof(float);
  vignet_kernel<<<1024, NT, shmem, stream>>>(P, (float*)d_out);
}